// GCN_HL01_bn_relu_42545946034233
// MI455X (gfx1250) — compile-verified
//
#include <hip/hip_runtime.h>

#define NN 50000
#define NE 800000
#define F 128
#define BN_EPS 1e-5f
#define RRELU_SLOPE (11.0f / 48.0f)

typedef __attribute__((ext_vector_type(2))) float v2f;
typedef __attribute__((ext_vector_type(8))) float v8f;

// ---------------------------------------------------------------- zero
__global__ void zero_kernel(float* __restrict__ p, int n) {
  int i = blockIdx.x * blockDim.x + threadIdx.x;
  int stride = gridDim.x * blockDim.x;
  for (; i < n; i += stride) p[i] = 0.0f;
}

// ---------------------------------------------------------------- scatter
// One wave32 per edge: lane handles 4 features (float4 load, 4 f32 atomics).
__global__ void scatter_kernel(const float* __restrict__ x,
                               const int* __restrict__ src,
                               const int* __restrict__ tgt,
                               const float* __restrict__ ew,
                               float* __restrict__ agg, int E) {
  int gwave = (blockIdx.x * blockDim.x + threadIdx.x) >> 5;
  int lane = threadIdx.x & 31;
  int nwaves = (gridDim.x * blockDim.x) >> 5;
  for (int e = gwave; e < E; e += nwaves) {
    int s = src[e];
    int t = tgt[e];
    float w = ew[e];
    const float4* xr = (const float4*)(x + (size_t)s * F);
    float4 v = xr[lane];
    float* ar = agg + (size_t)t * F + lane * 4;
    atomicAdd(ar + 0, w * v.x);
    atomicAdd(ar + 1, w * v.y);
    atomicAdd(ar + 2, w * v.z);
    atomicAdd(ar + 3, w * v.w);
  }
}

// ---------------------------------------------------------------- WMMA GEMM
// out[m][n] = sum_k A1[m][k]*Wrel[n][k] + A2[m][k]*Wroot[n][k] + bias[n]
// Block = 256 threads = 8 waves. Block handles one 16-row tile; wave w handles
// column tile n0 = 16*w. A-tiles staged in LDS (shared by all 8 waves).
// V_WMMA_F32_16X16X4_F32 layouts (ISA 7.12.2):
//   A 16x4:  lane<16: M=lane, K={k0,k0+1}; lane>=16: M=lane-16, K={k0+2,k0+3}
//   B  4x16: mirrored (lane half selects K-pair, lane&15 = N)
//   C/D:     VGPR v: lanes<16 -> M=v, N=lane; lanes>=16 -> M=v+8, N=lane-16
__global__ void gemm_dual_wmma(const float* __restrict__ A1,
                               const float* __restrict__ Wrel,
                               const float* __restrict__ A2,
                               const float* __restrict__ Wroot,
                               const float* __restrict__ bias,
                               float* __restrict__ out) {
  __shared__ float lA1[16 * F];
  __shared__ float lA2[16 * F];

  const int m0 = blockIdx.x * 16;

  // Stage both 16x128 A-tiles into LDS (coalesced b32 loads).
  for (int i = threadIdx.x; i < 16 * F; i += blockDim.x) {
    int r = i >> 7;
    int c = i & (F - 1);
    lA1[i] = A1[(size_t)(m0 + r) * F + c];
    lA2[i] = A2[(size_t)(m0 + r) * F + c];
  }
  __syncthreads();

  const int wave = threadIdx.x >> 5;  // 0..7 -> column tile
  const int lane = threadIdx.x & 31;
  const int half = lane >> 4;         // selects K-pair within the 4-wide step
  const int l16 = lane & 15;
  const int n0 = wave * 16;

  v8f acc = {};

#pragma unroll 8
  for (int k0 = 0; k0 < F; k0 += 4) {
    const int ka = k0 + 2 * half;

    // A (rel path) from LDS: row m0+l16, cols ka..ka+1  (ds_load_b64)
    v2f a_rel = *(const v2f*)(&lA1[l16 * F + ka]);
    // B[k][n] = Wrel[n][k]; lane holds n = n0+l16, k = ka..ka+1 (row-major contiguous)
    v2f b_rel = *(const v2f*)(Wrel + (size_t)(n0 + l16) * F + ka);
    acc = __builtin_amdgcn_wmma_f32_16x16x4_f32(false, a_rel, false, b_rel,
                                                (short)0, acc, false, false);

    v2f a_root = *(const v2f*)(&lA2[l16 * F + ka]);
    v2f b_root = *(const v2f*)(Wroot + (size_t)(n0 + l16) * F + ka);
    acc = __builtin_amdgcn_wmma_f32_16x16x4_f32(false, a_root, false, b_root,
                                                (short)0, acc, false, false);
  }

  const float b = bias[n0 + l16];
#pragma unroll
  for (int v = 0; v < 8; ++v) {
    int row = m0 + v + 8 * half;
    out[(size_t)row * F + n0 + l16] = acc[v] + b;
  }
}

// ---------------------------------------------------------------- BN stats
// stride is a multiple of 128 so each thread stays on one column.
__global__ void bn_stats_kernel(const float* __restrict__ h,
                                float* __restrict__ sums,
                                float* __restrict__ sumsq, int total) {
  int tid = blockIdx.x * blockDim.x + threadIdx.x;
  int stride = gridDim.x * blockDim.x;
  int col = tid & (F - 1);
  float s = 0.0f, sq = 0.0f;
  for (int i = tid; i < total; i += stride) {
    float v = h[i];
    s += v;
    sq += v * v;
  }
  atomicAdd(&sums[col], s);
  atomicAdd(&sumsq[col], sq);
}

// ---------------------------------------------------------------- BN apply + RReLU
__global__ void bn_apply_kernel(float* __restrict__ h,
                                const float* __restrict__ sums,
                                const float* __restrict__ sumsq,
                                const float* __restrict__ gamma,
                                const float* __restrict__ beta, int total) {
  const float invN = 1.0f / (float)NN;
  int tid = blockIdx.x * blockDim.x + threadIdx.x;
  int stride = gridDim.x * blockDim.x;
  for (int i = tid; i < total; i += stride) {
    int col = i & (F - 1);
    float mean = sums[col] * invN;
    float var = sumsq[col] * invN - mean * mean;
    float v = (h[i] - mean) * rsqrtf(var + BN_EPS) * gamma[col] + beta[col];
    h[i] = (v >= 0.0f) ? v : v * RRELU_SLOPE;
  }
}

// ---------------------------------------------------------------- launch
extern "C" void kernel_launch(void* const* d_in, const int* in_sizes, int n_in,
                              void* d_out, int out_size, void* d_ws, size_t ws_size,
                              hipStream_t stream) {
  const float* x       = (const float*)d_in[0];
  const int*   eidx    = (const int*)d_in[1];
  const float* eattr   = (const float*)d_in[2];
  const float* W1_rel  = (const float*)d_in[3];
  const float* b1      = (const float*)d_in[4];
  const float* W1_root = (const float*)d_in[5];
  const float* gamma   = (const float*)d_in[6];
  const float* beta    = (const float*)d_in[7];
  const float* W2_rel  = (const float*)d_in[8];
  const float* b2      = (const float*)d_in[9];
  const float* W2_root = (const float*)d_in[10];

  const int* src = eidx;        // edge_index[0]
  const int* tgt = eidx + NE;   // edge_index[1]

  float* agg   = (float*)d_ws;               // NN*F
  float* h     = agg + (size_t)NN * F;       // NN*F
  float* sums  = h + (size_t)NN * F;         // F
  float* sumsq = sums + F;                   // F
  float* out   = (float*)d_out;

  const int total = NN * F;
  const int scatterBlocks = (NE * 32 + 255) / 256;  // one wave per edge

  // ---- layer 1 ----
  zero_kernel<<<2048, 256, 0, stream>>>(agg, total);
  zero_kernel<<<1, 256, 0, stream>>>(sums, 2 * F);
  scatter_kernel<<<scatterBlocks, 256, 0, stream>>>(x, src, tgt, eattr, agg, NE);
  gemm_dual_wmma<<<NN / 16, 256, 0, stream>>>(agg, W1_rel, x, W1_root, b1, h);
  bn_stats_kernel<<<1024, 256, 0, stream>>>(h, sums, sumsq, total);
  bn_apply_kernel<<<2048, 256, 0, stream>>>(h, sums, sumsq, gamma, beta, total);

  // ---- layer 2 ----
  zero_kernel<<<2048, 256, 0, stream>>>(agg, total);
  scatter_kernel<<<scatterBlocks, 256, 0, stream>>>(h, src, tgt, eattr, agg, NE);
  gemm_dual_wmma<<<NN / 16, 256, 0, stream>>>(agg, W2_rel, h, W2_root, b2, out);
}